// Retrace_7524782703163
// MI455X (gfx1250) — compile-verified
//
#include <hip/hip_runtime.h>
#include <hip/hip_bf16.h>
#include <stdint.h>

// Problem constants (match reference)
#define T1      2049          // rows of every input
#define NT      2048          // rows compared in the loss
#define BCOLS   8192          // batch (and num_actions)
#define GAMMA   0.99f
#define INV_NA  (1.0f / 8192.0f)

// Chunked-scan configuration
#define NCHUNK  64            // parallel chunks over the time axis
#define CHUNK   32            // steps per chunk (NCHUNK*CHUNK >= 2047)
#define WROWS   4             // t-rows staged per LDS window
#define NWIN    (CHUNK / WROWS)         // 8 windows per chunk
#define COLS    256           // batch columns per block (== blockDim.x)
#define BUF_FLOATS (5 * WROWS * COLS)   // one staging buffer: 5 arrays x 4 rows x 256 cols
#define NBUF    3             // triple buffer: 2 async stages in flight

// ---- gfx1250 async global->LDS path (guarded; falls back to ds stores) ----
#if defined(__HIP_DEVICE_COMPILE__) && defined(__gfx1250__) && \
    __has_builtin(__builtin_amdgcn_global_load_async_to_lds_b128) && \
    __has_builtin(__builtin_amdgcn_s_wait_asynccnt)
#define USE_ASYNC 1
typedef int v4i_ __attribute__((vector_size(4 * sizeof(int))));
typedef __attribute__((address_space(1))) v4i_ gv4i_t;   // global int4
typedef __attribute__((address_space(3))) v4i_ lv4i_t;   // LDS int4
#define WAIT_ASYNC_LE(n) __builtin_amdgcn_s_wait_asynccnt(n)
#else
#define USE_ASYNC 0
#define WAIT_ASYNC_LE(n)
#endif

// Stage one window: rows [tw_lo, tw_lo+WROWS) of {rewards(row t), eQ,tQ,tp,bp (row t+1)}
// into lds_buf. 5 arrays * 4 rows * 256 cols * 4B = 20KB => 5 x async b128 per thread.
__device__ __forceinline__ void stage_window(
    const float* __restrict__ rw, const float* __restrict__ eQ,
    const float* __restrict__ tQ, const float* __restrict__ tp,
    const float* __restrict__ bp,
    float* __restrict__ lds_buf, int tw_lo, int c0, int tid)
{
#pragma unroll
  for (int jj = 0; jj < 5; ++jj) {
    int f  = tid + jj * 256;          // 0..1279  (flat 16B-unit index)
    int u  = f & 63;                  // float4 unit within a 256-col row
    int iw = f >> 6;                  // 0..19 == i*WROWS + w
    int w  = iw & (WROWS - 1);
    int i  = iw >> 2;                 // 0..4 : which array
    const float* base; int rowoff;
    if      (i == 0) { base = rw; rowoff = 0; }
    else if (i == 1) { base = eQ; rowoff = 1; }
    else if (i == 2) { base = tQ; rowoff = 1; }
    else if (i == 3) { base = tp; rowoff = 1; }
    else             { base = bp; rowoff = 1; }
    const float* g = base + (size_t)(tw_lo + w + rowoff) * BCOLS + c0 + (u << 2);
    float* l = lds_buf + (f << 2);    // layout: [i][w][col], 16B aligned
#if USE_ASYNC
    __builtin_amdgcn_global_load_async_to_lds_b128(
        (gv4i_t*)(void*)g, (lv4i_t*)(void*)l, 0, 0);
#else
    *(float4*)l = *(const float4*)g;
#endif
  }
}

// PASS==1: compose chunk affine map (A,M) with G[lo] = A + M*G[hi]; write wsA/wsM.
// PASS==3: start from boundary G[hi] (Hs), rebuild G[t], accumulate (Q[t-1]-G[t])^2.
template <int PASS>
__global__ void __launch_bounds__(256) retrace_scan_kernel(
    const float* __restrict__ Q,  const float* __restrict__ eQ,
    const float* __restrict__ tQ, const float* __restrict__ rw,
    const float* __restrict__ tp, const float* __restrict__ bp,
    const float* __restrict__ Hs, float* __restrict__ wsA,
    float* __restrict__ wsM, float* __restrict__ partials)
{
  __shared__ __align__(16) float lds[NBUF * BUF_FLOATS];  // 60KB triple buffer
  __shared__ float red[256];
  const int tid = threadIdx.x;
  const int k   = blockIdx.y;
  const int c0  = blockIdx.x * COLS;
  const int b   = c0 + tid;
  const int lo  = 1 + k * CHUNK;
  const int hi  = (lo + CHUNK < NT) ? (lo + CHUNK) : NT;   // k==63 -> 2048 (31 steps)

  float A = 0.0f, M = 1.0f;    // PASS 1 state
  float G = 0.0f, acc = 0.0f;  // PASS 3 state
  if (PASS == 3) {
    G = Hs[k * BCOLS + b];
    if (k == NCHUNK - 1) {     // boundary term: (Q[2047] - G[2048])^2
      float d = Q[(size_t)(NT - 1) * BCOLS + b] - G;
      acc = d * d;
    }
  }

  // Prologue: windows 0 and 1 in flight (window n stages rows [hi - W*(n+1), hi - W*n)).
  stage_window(rw, eQ, tQ, tp, bp, lds + 0 * BUF_FLOATS, hi - WROWS,     c0, tid);
  stage_window(rw, eQ, tQ, tp, bp, lds + 1 * BUF_FLOATS, hi - 2 * WROWS, c0, tid);

  int cur = 0;                 // buffer holding window j
#pragma unroll 1
  for (int j = 0; j < NWIN; ++j) {
    const int tw_lo = hi - WROWS * (j + 1);
    // Oldest stage (window j) complete; window j+1 may stay in flight (5 instrs).
    if (j + 1 < NWIN) { WAIT_ASYNC_LE(5); } else { WAIT_ASYNC_LE(0); }
    __syncthreads();
    if (j + 2 < NWIN) {
      int nb = cur + 2; if (nb >= NBUF) nb -= NBUF;
      stage_window(rw, eQ, tQ, tp, bp, lds + nb * BUF_FLOATS,
                   hi - WROWS * (j + 3), c0, tid);
      if (PASS == 3) {
#pragma unroll
        for (int w2 = 0; w2 < WROWS; ++w2)   // gfx1250 global_prefetch_b8, 2 windows ahead
          __builtin_prefetch(Q + (size_t)(hi - WROWS * (j + 3) - 1 + w2) * BCOLS + b, 0, 1);
      }
    }
    const float* Lb = lds + cur * BUF_FLOATS;
#pragma unroll
    for (int w = WROWS - 1; w >= 0; --w) {
      int t = tw_lo + w;
      if (t < lo) continue;                  // uniform: only last window of k==63
      float rv = Lb[(0 * WROWS + w) * COLS + tid];
      float eq = Lb[(1 * WROWS + w) * COLS + tid];
      float tq = Lb[(2 * WROWS + w) * COLS + tid];
      float dp = Lb[(3 * WROWS + w) * COLS + tid] - Lb[(4 * WROWS + w) * COLS + tid];
      float c  = __expf(fminf(dp, 0.0f) * INV_NA);
      float m  = GAMMA * c;
      float a  = fmaf(GAMMA, eq, rv) - m * tq;   // r + g*(eq - c*tq)
      if (PASS == 1) {
        A = fmaf(m, A, a);
        M = m * M;
      } else {
        G = fmaf(m, G, a);
        float d = Q[(size_t)(t - 1) * BCOLS + b] - G;
        acc = fmaf(d, d, acc);
      }
    }
    cur = (cur + 1 == NBUF) ? 0 : cur + 1;
  }

  if (PASS == 1) {
    wsA[k * BCOLS + b] = A;
    wsM[k * BCOLS + b] = M;
  } else {
    red[tid] = acc;
    __syncthreads();
#pragma unroll
    for (int s = 128; s > 0; s >>= 1) {
      if (tid < s) red[tid] += red[tid + s];
      __syncthreads();
    }
    if (tid == 0) partials[k * gridDim.x + blockIdx.x] = red[0];
  }
}

// Pass 2: serial combine over chunk boundaries. Hs[k][b] = G at t == hi_k.
__global__ void __launch_bounds__(256) retrace_combine_kernel(
    const float* __restrict__ tQ, const float* __restrict__ wsA,
    const float* __restrict__ wsM, float* __restrict__ Hs)
{
  const int b = blockIdx.x * blockDim.x + threadIdx.x;
  float G = tQ[(size_t)NT * BCOLS + b];          // G[2048] = target_Q[2048][b]
#pragma unroll 4
  for (int k = NCHUNK - 1; k >= 0; --k) {
    Hs[k * BCOLS + b] = G;
    G = fmaf(wsM[k * BCOLS + b], G, wsA[k * BCOLS + b]);
  }
}

// Pass 4: deterministic fixed-order reduction of block partials -> mean.
__global__ void __launch_bounds__(256) retrace_finalize_kernel(
    const float* __restrict__ partials, int n, float* __restrict__ out)
{
  __shared__ float red[256];
  float s = 0.0f;
  for (int i = threadIdx.x; i < n; i += 256) s += partials[i];
  red[threadIdx.x] = s;
  __syncthreads();
#pragma unroll
  for (int st = 128; st > 0; st >>= 1) {
    if (threadIdx.x < st) red[threadIdx.x] += red[threadIdx.x + st];
    __syncthreads();
  }
  if (threadIdx.x == 0) out[0] = red[0] * (1.0f / 16777216.0f);  // 1/(2048*8192)
}

extern "C" void kernel_launch(void* const* d_in, const int* in_sizes, int n_in,
                              void* d_out, int out_size, void* d_ws, size_t ws_size,
                              hipStream_t stream) {
  const float* Q  = (const float*)d_in[0];
  const float* eQ = (const float*)d_in[1];
  const float* tQ = (const float*)d_in[2];
  const float* rw = (const float*)d_in[3];
  const float* tp = (const float*)d_in[4];
  const float* bp = (const float*)d_in[5];
  float* out = (float*)d_out;

  // Workspace layout (~6.3 MB): wsA | wsM | Hs | partials
  float* wsA = (float*)d_ws;
  float* wsM = wsA + (size_t)NCHUNK * BCOLS;
  float* Hs  = wsM + (size_t)NCHUNK * BCOLS;
  float* partials = Hs + (size_t)NCHUNK * BCOLS;

  dim3 grid(BCOLS / COLS, NCHUNK);   // (32, 64)
  dim3 blk(256);
  retrace_scan_kernel<1><<<grid, blk, 0, stream>>>(Q, eQ, tQ, rw, tp, bp,
                                                   nullptr, wsA, wsM, nullptr);
  retrace_combine_kernel<<<dim3(BCOLS / 256), blk, 0, stream>>>(tQ, wsA, wsM, Hs);
  retrace_scan_kernel<3><<<grid, blk, 0, stream>>>(Q, eQ, tQ, rw, tp, bp,
                                                   Hs, nullptr, nullptr, partials);
  retrace_finalize_kernel<<<dim3(1), blk, 0, stream>>>(partials,
                                                       NCHUNK * (BCOLS / COLS), out);
}